// MultiCrossEntropyLoss_14809047236868
// MI455X (gfx1250) — compile-verified
//
#include <hip/hip_runtime.h>
#include <hip/hip_bf16.h>

#define ROWS 2048
#define COLS 50257
#define NEG_BIG (-3.402823466e+38f)

typedef __attribute__((ext_vector_type(2))) float v2f;
typedef __attribute__((ext_vector_type(4))) float v4f;
typedef __attribute__((ext_vector_type(8))) float v8f;

// ---- online softmax scalar update: tracks mx,Sx,U (U = sum exp(x-mx)*y) and my,Sy ----
__device__ __forceinline__ void online_pair(float xv, float yv,
                                            float& mx, float& Sx, float& U,
                                            float& my, float& Sy) {
    if (xv > mx) { float sc = __expf(mx - xv); Sx *= sc; U *= sc; mx = xv; }
    float e = __expf(xv - mx);
    Sx += e;
    U  += e * yv;
    if (yv > my) { float sc = __expf(my - yv); Sy *= sc; my = yv; }
    Sy += __expf(yv - my);
}

// ---- online softmax update for a float4 group (one rescale per group, only if max grows) ----
__device__ __forceinline__ void online_quad(v4f xv, v4f yv,
                                            float& mx, float& Sx, float& U,
                                            float& my, float& Sy) {
    float vm = fmaxf(fmaxf(xv.x, xv.y), fmaxf(xv.z, xv.w));
    if (vm > mx) { float sc = __expf(mx - vm); Sx *= sc; U *= sc; mx = vm; }
    float e0 = __expf(xv.x - mx), e1 = __expf(xv.y - mx);
    float e2 = __expf(xv.z - mx), e3 = __expf(xv.w - mx);
    Sx += (e0 + e1) + (e2 + e3);
    U  += (e0 * yv.x + e1 * yv.y) + (e2 * yv.z + e3 * yv.w);

    float wm = fmaxf(fmaxf(yv.x, yv.y), fmaxf(yv.z, yv.w));
    if (wm > my) { float sc = __expf(my - wm); Sy *= sc; my = wm; }
    Sy += (__expf(yv.x - my) + __expf(yv.y - my)) +
          (__expf(yv.z - my) + __expf(yv.w - my));
}

__device__ __forceinline__ void combine5(float& mx, float& Sx, float& U,
                                         float& my, float& Sy,
                                         float m2, float s2, float u2,
                                         float my2, float sy2) {
    float nm = fmaxf(mx, m2);
    float a = __expf(mx - nm), b = __expf(m2 - nm);
    Sx = Sx * a + s2 * b;
    U  = U  * a + u2 * b;
    mx = nm;
    float nmy = fmaxf(my, my2);
    Sy = Sy * __expf(my - nmy) + sy2 * __expf(my2 - nmy);
    my = nmy;
}

// One block (512 threads = 16 wave32) per row; streaming NT b128 loads.
__global__ void ce_rows_kernel(const float* __restrict__ x,
                               const float* __restrict__ y,
                               float* __restrict__ part) {
    const int row  = blockIdx.x;
    const int tid  = threadIdx.x;
    const int nthr = blockDim.x;

    const float* xr = x + (size_t)row * COLS;
    const float* yr = y + (size_t)row * COLS;

    float mx = NEG_BIG, Sx = 0.0f, U = 0.0f;
    float my = NEG_BIG, Sy = 0.0f;

    // Peel to 16B alignment: (row*COLS) % 4 == row % 4 since COLS % 4 == 1.
    const int p = (4 - (row & 3)) & 3;
    if (tid < p) online_pair(xr[tid], yr[tid], mx, Sx, U, my, Sy);

    const int n4 = (COLS - p) >> 2;
    const v4f* x4 = (const v4f*)(xr + p);
    const v4f* y4 = (const v4f*)(yr + p);
    for (int i = tid; i < n4; i += nthr) {
        v4f xv = __builtin_nontemporal_load(x4 + i);  // global_load_b128, NT (no reuse)
        v4f yv = __builtin_nontemporal_load(y4 + i);
        online_quad(xv, yv, mx, Sx, U, my, Sy);
    }

    const int done = p + (n4 << 2);
    for (int j = done + tid; j < COLS; j += nthr)
        online_pair(xr[j], yr[j], mx, Sx, U, my, Sy);

    // intra-wave tree reduction (wave32)
    for (int off = 16; off; off >>= 1) {
        float m2  = __shfl_down(mx, off, 32);
        float s2  = __shfl_down(Sx, off, 32);
        float u2  = __shfl_down(U,  off, 32);
        float my2 = __shfl_down(my, off, 32);
        float sy2 = __shfl_down(Sy, off, 32);
        combine5(mx, Sx, U, my, Sy, m2, s2, u2, my2, sy2);
    }

    __shared__ float sm[16], ss[16], su[16], tm[16], ts[16];
    const int wave = tid >> 5, lane = tid & 31;
    if (lane == 0) { sm[wave] = mx; ss[wave] = Sx; su[wave] = U; tm[wave] = my; ts[wave] = Sy; }
    __syncthreads();

    if (wave == 0) {
        if (lane < 16) { mx = sm[lane]; Sx = ss[lane]; U = su[lane]; my = tm[lane]; Sy = ts[lane]; }
        else           { mx = NEG_BIG;  Sx = 0.0f;     U = 0.0f;     my = NEG_BIG;  Sy = 0.0f;    }
        for (int off = 16; off; off >>= 1) {
            float m2  = __shfl_down(mx, off, 32);
            float s2  = __shfl_down(Sx, off, 32);
            float u2  = __shfl_down(U,  off, 32);
            float my2 = __shfl_down(my, off, 32);
            float sy2 = __shfl_down(Sy, off, 32);
            combine5(mx, Sx, U, my, Sy, m2, s2, u2, my2, sy2);
        }
        if (lane == 0) part[row] = U / Sx - my - __logf(Sy);
    }
}

// Final reduction of 2048 per-row dots via V_WMMA_F32_16X16X4_F32 (full fp32).
// A = 16x4 tile of partials, B = all-ones 4x16, C/D accumulates row-sums.
// One wave32, EXEC all ones, no divergence around the WMMA.
__global__ void ce_final_kernel(const float* __restrict__ part,
                                float* __restrict__ out) {
    const int lane = threadIdx.x;        // 0..31
    const int m  = lane & 15;            // A-matrix row this lane holds
    const int kh = lane >> 4;            // lanes 0-15: K=0,1; lanes 16-31: K=2,3
    v2f b; b.x = 1.0f; b.y = 1.0f;
    v8f acc = {};
    for (int i = 0; i < ROWS; i += 64) {
        const float* pp = part + i + m * 4 + kh * 2;
        v2f a; a.x = pp[0]; a.y = pp[1];
        acc = __builtin_amdgcn_wmma_f32_16x16x4_f32(
            /*neg_a=*/false, a, /*neg_b=*/false, b,
            /*c_mod=*/(short)0, acc, /*reuse_a=*/false, /*reuse_b=*/false);
    }
    // Every D element D[m][n] = rowsum_m (replicated across 16 columns) ->
    // summing all lanes/VGPRs gives 16x the true total.
    float s = ((acc[0] + acc[1]) + (acc[2] + acc[3])) +
              ((acc[4] + acc[5]) + (acc[6] + acc[7]));
    for (int off = 16; off; off >>= 1) s += __shfl_down(s, off, 32);
    if (lane == 0) out[0] = -(s * (1.0f / 16.0f)) * (1.0f / (float)ROWS);
}

extern "C" void kernel_launch(void* const* d_in, const int* in_sizes, int n_in,
                              void* d_out, int out_size, void* d_ws, size_t ws_size,
                              hipStream_t stream) {
    (void)in_sizes; (void)n_in; (void)out_size; (void)ws_size;
    const float* x = (const float*)d_in[0];
    const float* y = (const float*)d_in[1];
    float* out  = (float*)d_out;
    float* part = (float*)d_ws;          // 2048 floats of scratch

    ce_rows_kernel<<<dim3(ROWS), dim3(512), 0, stream>>>(x, y, part);
    ce_final_kernel<<<dim3(1), dim3(32), 0, stream>>>(part, out);
}